// GCNModelVAE_13743895347838
// MI455X (gfx1250) — compile-verified
//
#include <hip/hip_runtime.h>
#include <hip/hip_bf16.h>

typedef float v2f __attribute__((ext_vector_type(2)));
typedef float v8f __attribute__((ext_vector_type(8)));

#define N_NODES 8192
#define F_IN    512
#define H1_DIM  256
#define H2_DIM  64
#define N_EDGES 524288

// ---------------------------------------------------------------------------
// Dense GEMM: C[M,N] = op(A)[M,K] @ B[K,N], all f32 row-major.
// One 16x16 output tile per wave32; K stepped by 4 via V_WMMA_F32_16X16X4_F32.
// A-frag: lane l<16 holds M=l, K={2hi,2hi+1} in 2 VGPRs (contiguous float2).
// B-frag: symmetric layout, lane half selects K rows {2hi,2hi+1}, N=l.
// C/D: VGPR r -> rows (r, r+8), N striped over lane halves.
// ---------------------------------------------------------------------------
template <bool RELU_A>
__global__ void gemm_wmma_f32(const float* __restrict__ A,
                              const float* __restrict__ B,
                              float* __restrict__ C,
                              int M, int K, int N) {
    const int lane  = threadIdx.x & 31;
    const int wave  = threadIdx.x >> 5;
    const int tile  = blockIdx.x * (blockDim.x >> 5) + wave;   // wave-uniform
    const int tilesN = N >> 4;
    const int numTiles = (M >> 4) * tilesN;
    if (tile >= numTiles) return;                               // uniform exit
    const int tm = (tile / tilesN) << 4;
    const int tn = (tile % tilesN) << 4;
    const int l  = lane & 15;
    const int hi = lane >> 4;

    const float* __restrict__ arow = A + (size_t)(tm + l) * K + 2 * hi;
    const float* __restrict__ bcol = B + tn + l;

    v8f c = {};
    for (int k = 0; k < K; k += 4) {
        v2f a = *(const v2f*)(arow + k);
        if (RELU_A) { a.x = fmaxf(a.x, 0.0f); a.y = fmaxf(a.y, 0.0f); }
        v2f b;
        b.x = bcol[(size_t)(k + 2 * hi) * N];
        b.y = bcol[(size_t)(k + 2 * hi + 1) * N];
        c = __builtin_amdgcn_wmma_f32_16x16x4_f32(false, a, false, b,
                                                  (short)0, c, false, false);
    }

    float* __restrict__ crow = C + (size_t)(tm + hi * 8) * N + tn + l;
#pragma unroll
    for (int r = 0; r < 8; ++r) crow[(size_t)r * N] = c[r];
}

// ---------------------------------------------------------------------------
// Decoder: C[M,M] = Z @ Z^T, Z[M,K] row-major. Both fragments are contiguous
// float2 loads from Z (A: row tm+l, B: row tn+l). Output is 256 MB of
// write-once data -> non-temporal stores (TH=NT) so it does not evict the
// L2-resident working set; HBM write bandwidth is the stage's roofline.
// ---------------------------------------------------------------------------
__global__ void gemm_nt_wmma_f32(const float* __restrict__ Z,
                                 float* __restrict__ C,
                                 int M, int K) {
    const int  lane   = threadIdx.x & 31;
    const int  wave   = threadIdx.x >> 5;
    const long tile   = (long)blockIdx.x * (blockDim.x >> 5) + wave;
    const int  tilesN = M >> 4;
    const long numTiles = (long)tilesN * tilesN;
    if (tile >= numTiles) return;                               // uniform exit
    const int tm = (int)(tile / tilesN) << 4;
    const int tn = (int)(tile % tilesN) << 4;
    const int l  = lane & 15;
    const int hi = lane >> 4;

    const float* __restrict__ arow = Z + (size_t)(tm + l) * K + 2 * hi;
    const float* __restrict__ brow = Z + (size_t)(tn + l) * K + 2 * hi;

    v8f c = {};
#pragma unroll 4
    for (int k = 0; k < K; k += 4) {
        v2f a = *(const v2f*)(arow + k);
        v2f b = *(const v2f*)(brow + k);
        c = __builtin_amdgcn_wmma_f32_16x16x4_f32(false, a, false, b,
                                                  (short)0, c, false, false);
    }

    float* __restrict__ crow = C + (size_t)(tm + hi * 8) * M + tn + l;
#pragma unroll
    for (int r = 0; r < 8; ++r)
        __builtin_nontemporal_store(c[r], crow + (size_t)r * M);
}

// ---------------------------------------------------------------------------
// SpMM scatter: out[dst[e],:] += w[e] * h[src[e],:]. One (edge, 4-col) pair
// per thread; consecutive lanes cover consecutive columns of the same edge
// (coalesced float4 gather, 4 f32 atomics to L2). D4 = D/4.
// ---------------------------------------------------------------------------
__global__ void spmm_scatter4(const int* __restrict__ src,
                              const int* __restrict__ dst,
                              const float* __restrict__ w,
                              const float* __restrict__ h,
                              float* __restrict__ out,
                              int D4) {
    const long tid   = (long)blockIdx.x * blockDim.x + threadIdx.x;
    const long total = (long)N_EDGES * D4;
    if (tid >= total) return;
    const int e = (int)(tid / D4);
    const int c = (int)(tid % D4) * 4;
    const int D = D4 * 4;

    const float wt = w[e];
    const float4 v = *(const float4*)(h + (size_t)src[e] * D + c);
    float* o = out + (size_t)dst[e] * D + c;
    atomicAdd(o + 0, v.x * wt);
    atomicAdd(o + 1, v.y * wt);
    atomicAdd(o + 2, v.z * wt);
    atomicAdd(o + 3, v.w * wt);
}

// ---------------------------------------------------------------------------
// Reparameterization: z = z_mean + eps * exp(z_log_std)
// ---------------------------------------------------------------------------
__global__ void reparam_kernel(const float* __restrict__ zm,
                               const float* __restrict__ zl,
                               const float* __restrict__ eps,
                               float* __restrict__ z, int n) {
    const int i = blockIdx.x * blockDim.x + threadIdx.x;
    if (i < n) z[i] = zm[i] + eps[i] * expf(zl[i]);
}

extern "C" void kernel_launch(void* const* d_in, const int* in_sizes, int n_in,
                              void* d_out, int out_size, void* d_ws, size_t ws_size,
                              hipStream_t stream) {
    const float* x    = (const float*)d_in[0];
    const int*   esrc = (const int*)d_in[1];
    const int*   edst = (const int*)d_in[2];
    const float* ew   = (const float*)d_in[3];
    const float* eps  = (const float*)d_in[4];
    const float* W1   = (const float*)d_in[5];
    const float* W2   = (const float*)d_in[6];
    const float* W3   = (const float*)d_in[7];
    float* out = (float*)d_out;

    // Workspace layout (floats)
    float* ws = (float*)d_ws;
    float* t0 = ws;                                   // [N, H1] x@W1
    float* h1 = t0 + (size_t)N_NODES * H1_DIM;        // [N, H1] spmm(t0)
    float* t2 = h1 + (size_t)N_NODES * H1_DIM;        // [N, H2] relu(h1)@W2
    float* t3 = t2 + (size_t)N_NODES * H2_DIM;        // [N, H2] relu(h1)@W3
    float* zm = t3 + (size_t)N_NODES * H2_DIM;        // [N, H2] spmm(t2)
    float* zl = zm + (size_t)N_NODES * H2_DIM;        // [N, H2] spmm(t3)
    float* z  = zl + (size_t)N_NODES * H2_DIM;        // [N, H2]

    const dim3 blk(256);                              // 8 waves / block

    // 1) t0 = x @ W1
    {
        const int tiles = (N_NODES / 16) * (H1_DIM / 16);     // 8192
        gemm_wmma_f32<false><<<tiles / 8, blk, 0, stream>>>(
            x, W1, t0, N_NODES, F_IN, H1_DIM);
    }

    // 2) h1 = spmm(t0)   (pre-ReLU; ReLU fused into next GEMM's A-load)
    hipMemsetAsync(h1, 0, (size_t)N_NODES * H1_DIM * sizeof(float), stream);
    {
        const long total = (long)N_EDGES * (H1_DIM / 4);
        spmm_scatter4<<<(int)((total + 255) / 256), blk, 0, stream>>>(
            esrc, edst, ew, t0, h1, H1_DIM / 4);
    }

    // 3) t2 = relu(h1) @ W2 ; t3 = relu(h1) @ W3
    {
        const int tiles = (N_NODES / 16) * (H2_DIM / 16);     // 2048
        gemm_wmma_f32<true><<<tiles / 8, blk, 0, stream>>>(
            h1, W2, t2, N_NODES, H1_DIM, H2_DIM);
        gemm_wmma_f32<true><<<tiles / 8, blk, 0, stream>>>(
            h1, W3, t3, N_NODES, H1_DIM, H2_DIM);
    }

    // 4) zm = spmm(t2) ; zl = spmm(t3)
    hipMemsetAsync(zm, 0, (size_t)N_NODES * H2_DIM * sizeof(float), stream);
    hipMemsetAsync(zl, 0, (size_t)N_NODES * H2_DIM * sizeof(float), stream);
    {
        const long total = (long)N_EDGES * (H2_DIM / 4);
        spmm_scatter4<<<(int)((total + 255) / 256), blk, 0, stream>>>(
            esrc, edst, ew, t2, zm, H2_DIM / 4);
        spmm_scatter4<<<(int)((total + 255) / 256), blk, 0, stream>>>(
            esrc, edst, ew, t3, zl, H2_DIM / 4);
    }

    // 5) z = zm + eps * exp(zl)
    {
        const int n = N_NODES * H2_DIM;
        reparam_kernel<<<(n + 255) / 256, blk, 0, stream>>>(zm, zl, eps, z, n);
    }

    // 6) out = z @ z^T   (HBM-bound: 256 MB non-temporal output)
    {
        const long tiles = (long)(N_NODES / 16) * (N_NODES / 16);  // 262144
        gemm_nt_wmma_f32<<<(int)(tiles / 8), blk, 0, stream>>>(
            z, out, N_NODES, H2_DIM);
    }
}